// HGT_OL_27951647162620
// MI455X (gfx1250) — compile-verified
//
#include <hip/hip_runtime.h>
#include <math.h>

// ---------------------------------------------------------------------------
// MI455X (gfx1250, wave32) implementation.
// GEMMs / attention / Gram products run on V_WMMA_F32_16X16X4_F32 (fp32
// matrix pipe, matching the fp32 reference numerics). Working set fits in the
// 192MB L2 so kernels are shaped for matrix-op throughput, not HBM.
// Global->LDS staging uses CDNA5 GLOBAL_LOAD_ASYNC_TO_LDS (ASYNCcnt) when the
// toolchain exposes the builtins, with a scalar fallback otherwise.
// Requires ws_size >= ~180 MB.
// ---------------------------------------------------------------------------

#define BB 4
#define NN 8192
#define DD 128
#define LEAF 32
#define HEADS 4
#define HD 32
#define SLOTS 24

#define OUT_STRIDE 3309568L   // per-batch output row: 262144 dense + 2*1523712 uv
#define US_BASE    262144L
#define VS_BASE    1785856L

typedef __attribute__((ext_vector_type(2))) float v2f;
typedef __attribute__((ext_vector_type(8))) float v8f;

__device__ __forceinline__ v8f wmma_f32(v2f a, v2f b, v8f c) {
  // D = A(16x4) * B(4x16) + C(16x16), fp32 matrix op
  return __builtin_amdgcn_wmma_f32_16x16x4_f32(false, a, false, b, (short)0, c,
                                               false, false);
}

// ---- CDNA5 async global->LDS path (guarded so compile never breaks) -------
#if defined(__has_builtin)
#if __has_builtin(__builtin_amdgcn_global_load_async_to_lds_b64) && \
    __has_builtin(__builtin_amdgcn_s_wait_asynccnt)
#define USE_ASYNC_LDS 1
#endif
#endif
#ifndef USE_ASYNC_LDS
#define USE_ASYNC_LDS 0
#endif

#if USE_ASYNC_LDS
// Builtin signature (from clang diagnostic): param0 = v2i addrspace(1)*,
// param1 = v2i addrspace(3)*, then two constant ints (offset, cpol).
typedef int v2i_vec __attribute__((vector_size(2 * sizeof(int))));
typedef __attribute__((address_space(1))) v2i_vec glob_v2i;
typedef __attribute__((address_space(3))) v2i_vec lds_v2i;

__device__ __forceinline__ void async_g2l_b64(void* lds, const void* gmem) {
  // On gfx1250 the low 32 bits of a flat LDS address are the LDS offset.
  __builtin_amdgcn_global_load_async_to_lds_b64(
      (glob_v2i*)(uintptr_t)gmem,
      (lds_v2i*)(unsigned int)(uintptr_t)lds,
      0, 0);
}
#endif

// ---------------------------------------------------------------------------
// 1) Neighbor embedding: h_pre = cen + sum_slots active * (feat @ nbr_w + b)
//    one wave per token, lane covers 4 of the 128 output dims
// ---------------------------------------------------------------------------
__global__ __launch_bounds__(256) void embed_kernel(
    const float* __restrict__ x, const float* __restrict__ cen_w,
    const float* __restrict__ cen_b, const float* __restrict__ nbr_w,
    const float* __restrict__ nbr_b, float* __restrict__ hpre)
{
  __shared__ float sx[8][124];
  const int lane = threadIdx.x & 31;
  const int wv = threadIdx.x >> 5;
  const int tok = blockIdx.x * 8 + wv;
  const float* xr = x + (size_t)tok * 124;
  for (int i = lane; i < 124; i += 32) sx[wv][i] = xr[i];
  __syncthreads();
  float acc[4];
#pragma unroll
  for (int j = 0; j < 4; ++j) {
    const int d = lane + 32 * j;
    float a = cen_b[d];
#pragma unroll
    for (int f = 0; f < 4; ++f) a += sx[wv][f] * cen_w[f * DD + d];
    acc[j] = a;
  }
  for (int s = 0; s < SLOTS; ++s) {
    const int base = 4 + 5 * s;
    const float rel = sx[wv][base];
    const float p0 = sx[wv][base + 1] - sx[wv][0];
    const float p1 = sx[wv][base + 2] - sx[wv][1];
    const float p2 = sx[wv][base + 3] - sx[wv][2];
    const float w  = sx[wv][base + 4];
    if (w != 0.0f) {
#pragma unroll
      for (int j = 0; j < 4; ++j) {
        const int d = lane + 32 * j;
        acc[j] += nbr_b[d] + rel * nbr_w[0 * DD + d] + p0 * nbr_w[1 * DD + d] +
                  p1 * nbr_w[2 * DD + d] + p2 * nbr_w[3 * DD + d] +
                  w * nbr_w[4 * DD + d];
      }
    }
  }
  float* hr = hpre + (size_t)tok * DD;
#pragma unroll
  for (int j = 0; j < 4; ++j) hr[lane + 32 * j] = acc[j];
}

// ---------------------------------------------------------------------------
// 2) Generic WMMA GEMM: C[M,N] = act(A[M,K] @ W[K,N] + bias) + res, 64x64 C
//    tile per 128-thread block (4 waves); each wave owns a 32x32 sub-tile:
//    2 A-frags x 2 B-frags -> 4 WMMAs per k-step.
//    LDS staging via async global->LDS copies when available.
//    act: 0 = none, 1 = exact GELU. res (optional) has same [M,N] shape.
//    Assumes M % 64 == 0 and K % 16 == 0 (true for all call sites).
// ---------------------------------------------------------------------------
__global__ __launch_bounds__(128) void gemm_kernel(
    const float* __restrict__ A, const float* __restrict__ W,
    const float* __restrict__ bias, const float* __restrict__ res,
    float* __restrict__ C, int M, int K, int N, int act)
{
  __shared__ float As[64][18];   // stride 18: 8B-aligned pairs, no bank camp
  __shared__ float Bs[16][66];
  const int tid = threadIdx.x;
  const int lane = tid & 31;
  const int wv = tid >> 5;       // 0..3
  const int wr = wv >> 1;        // 0..1 row group (32 rows)
  const int wc = wv & 1;         // 0..1 col group (32 cols)
  const int blockRow = blockIdx.y * 64;
  const int blockCol = blockIdx.x * 64;
  const bool fullB = (blockCol + 64) <= N;
  v8f c00 = {}, c01 = {}, c10 = {}, c11 = {};
  const int rl = lane & 15;
  const int kh = (lane >> 4) << 1;
  const int ar0 = wr * 32 + rl, ar1 = ar0 + 16;
  const int bc0 = wc * 32 + rl, bc1 = bc0 + 16;

  for (int k0 = 0; k0 < K; k0 += 16) {
#if USE_ASYNC_LDS
    // A tile 64x16: 512 8-byte async copies (all 8B aligned: K,k0,c even)
#pragma unroll
    for (int i = tid; i < 512; i += 128) {
      const int r = i >> 3;
      const int c = (i & 7) << 1;
      async_g2l_b64(&As[r][c], A + (size_t)(blockRow + r) * K + k0 + c);
    }
    if (fullB) {
#pragma unroll
      for (int i = tid; i < 512; i += 128) {
        const int r = i >> 5;
        const int c = (i & 31) << 1;
        async_g2l_b64(&Bs[r][c], W + (size_t)(k0 + r) * N + blockCol + c);
      }
    } else {
      for (int i = tid; i < 16 * 64; i += 128) {
        const int r = i >> 6, c = i & 63;
        const int gc = blockCol + c;
        Bs[r][c] = (gc < N) ? W[(size_t)(k0 + r) * N + gc] : 0.0f;
      }
    }
    __builtin_amdgcn_s_wait_asynccnt(0);
#else
    for (int i = tid; i < 64 * 16; i += 128) {
      const int r = i >> 4, c = i & 15;
      As[r][c] = A[(size_t)(blockRow + r) * K + k0 + c];
    }
    for (int i = tid; i < 16 * 64; i += 128) {
      const int r = i >> 6, c = i & 63;
      const int gc = blockCol + c;
      Bs[r][c] = (gc < N) ? W[(size_t)(k0 + r) * N + gc] : 0.0f;
    }
#endif
    __syncthreads();
#pragma unroll
    for (int kk = 0; kk < 16; kk += 4) {
      const int kf = kk + kh;
      v2f a0, a1, b0, b1;
      a0.x = As[ar0][kf];     a0.y = As[ar0][kf + 1];
      a1.x = As[ar1][kf];     a1.y = As[ar1][kf + 1];
      b0.x = Bs[kf][bc0];     b0.y = Bs[kf + 1][bc0];
      b1.x = Bs[kf][bc1];     b1.y = Bs[kf + 1][bc1];
      c00 = wmma_f32(a0, b0, c00); c01 = wmma_f32(a0, b1, c01);
      c10 = wmma_f32(a1, b0, c10); c11 = wmma_f32(a1, b1, c11);
    }
    __syncthreads();
  }

  const int rowOff = (lane >> 4) << 3;
  const int col0 = blockCol + wc * 32 + rl;
  const int col1 = col0 + 16;
  const float bv0 = (bias && col0 < N) ? bias[col0] : 0.0f;
  const float bv1 = (bias && col1 < N) ? bias[col1] : 0.0f;
#pragma unroll
  for (int v = 0; v < 8; ++v) {
    const int row0 = blockRow + wr * 32 + rowOff + v;
    const int row1 = row0 + 16;
    float e00 = c00[v] + bv0, e01 = c01[v] + bv1;
    float e10 = c10[v] + bv0, e11 = c11[v] + bv1;
    if (act) {
      e00 = 0.5f * e00 * (1.0f + erff(e00 * 0.70710678118654752f));
      e01 = 0.5f * e01 * (1.0f + erff(e01 * 0.70710678118654752f));
      e10 = 0.5f * e10 * (1.0f + erff(e10 * 0.70710678118654752f));
      e11 = 0.5f * e11 * (1.0f + erff(e11 * 0.70710678118654752f));
    }
    if (col0 < N) {
      if (res) e00 += res[(size_t)row0 * N + col0];
      C[(size_t)row0 * N + col0] = e00;
      if (res) e10 += res[(size_t)row1 * N + col0];
      C[(size_t)row1 * N + col0] = e10;
    }
    if (col1 < N) {
      if (res) e01 += res[(size_t)row0 * N + col1];
      C[(size_t)row0 * N + col1] = e01;
      if (res) e11 += res[(size_t)row1 * N + col1];
      C[(size_t)row1 * N + col1] = e11;
    }
  }
}

// ---------------------------------------------------------------------------
// 3) LayerNorm over D=128: one wave per token, wave32 shuffle reductions
// ---------------------------------------------------------------------------
__global__ __launch_bounds__(256) void layernorm_kernel(
    const float* __restrict__ in, const float* __restrict__ g,
    const float* __restrict__ b, float* __restrict__ out, int M)
{
  const int lane = threadIdx.x & 31;
  const int wv = threadIdx.x >> 5;
  const int tok = blockIdx.x * 8 + wv;
  if (tok >= M) return;
  const float* r = in + (size_t)tok * DD;
  float v0 = r[lane], v1 = r[lane + 32], v2 = r[lane + 64], v3 = r[lane + 96];
  float s = v0 + v1 + v2 + v3;
#pragma unroll
  for (int o = 16; o > 0; o >>= 1) s += __shfl_xor(s, o, 32);
  const float mu = s * (1.0f / 128.0f);
  v0 -= mu; v1 -= mu; v2 -= mu; v3 -= mu;
  float q = v0 * v0 + v1 * v1 + v2 * v2 + v3 * v3;
#pragma unroll
  for (int o = 16; o > 0; o >>= 1) q += __shfl_xor(q, o, 32);
  const float inv = rsqrtf(q * (1.0f / 128.0f) + 1e-5f);
  float* w = out + (size_t)tok * DD;
  w[lane]      = v0 * inv * g[lane]      + b[lane];
  w[lane + 32] = v1 * inv * g[lane + 32] + b[lane + 32];
  w[lane + 64] = v2 * inv * g[lane + 64] + b[lane + 64];
  w[lane + 96] = v3 * inv * g[lane + 96] + b[lane + 96];
}

// ---------------------------------------------------------------------------
// 4) Block attention (LEAF=32, HD=32) with RoPE. One wave per (block, head);
//    S = QK^T and O = PV via 2x2 grids of 16x16 WMMA tiles, K=32.
//    sQ is reused to hold the softmax matrix P after S is computed.
// ---------------------------------------------------------------------------
__global__ __launch_bounds__(128) void attn_kernel(
    const float* __restrict__ qkv, float* __restrict__ o)
{
  __shared__ float sQ[4][32][33];
  __shared__ float sK[4][32][33];
  __shared__ float sV[4][32][33];
  const int lane = threadIdx.x & 31;
  const int h = threadIdx.x >> 5;
  const int rowBase = blockIdx.x * 32;
  const float* qr = qkv + (size_t)(rowBase + lane) * 384;
  for (int d = 0; d < 32; ++d) {
    sQ[h][lane][d] = qr[h * 32 + d];
    sK[h][lane][d] = qr[128 + h * 32 + d];
    sV[h][lane][d] = qr[256 + h * 32 + d];
  }
  // RoPE on own row (lane == token position t)
  const float t = (float)lane;
#pragma unroll
  for (int i = 0; i < 16; ++i) {
    const float f = __expf(-(float)i * (9.210340371976184f / 16.0f)); // 10000^(-i/16)
    float sn, cs;
    __sincosf(t * f, &sn, &cs);
    float a = sQ[h][lane][i], b2 = sQ[h][lane][i + 16];
    sQ[h][lane][i]      = a * cs - b2 * sn;
    sQ[h][lane][i + 16] = b2 * cs + a * sn;
    a = sK[h][lane][i]; b2 = sK[h][lane][i + 16];
    sK[h][lane][i]      = a * cs - b2 * sn;
    sK[h][lane][i + 16] = b2 * cs + a * sn;
  }
  __syncthreads();
  const int rl = lane & 15;
  const int kh = (lane >> 4) << 1;
  v8f s00 = {}, s01 = {}, s10 = {}, s11 = {};
#pragma unroll
  for (int kk = 0; kk < 32; kk += 4) {
    const int kf = kk + kh;
    v2f a0, a1, b0, b1;
    a0.x = sQ[h][rl][kf];        a0.y = sQ[h][rl][kf + 1];
    a1.x = sQ[h][rl + 16][kf];   a1.y = sQ[h][rl + 16][kf + 1];
    b0.x = sK[h][rl][kf];        b0.y = sK[h][rl][kf + 1];      // B = K^T
    b1.x = sK[h][rl + 16][kf];   b1.y = sK[h][rl + 16][kf + 1];
    s00 = wmma_f32(a0, b0, s00); s01 = wmma_f32(a0, b1, s01);
    s10 = wmma_f32(a1, b0, s10); s11 = wmma_f32(a1, b1, s11);
  }
  const float scl = 0.17677669529663687f;  // 32^-0.5
  const int rowOff = (lane >> 4) << 3;
#pragma unroll
  for (int v = 0; v < 8; ++v) {
    const int r = rowOff + v;
    sQ[h][r][rl]           = s00[v] * scl;
    sQ[h][r][rl + 16]      = s01[v] * scl;
    sQ[h][r + 16][rl]      = s10[v] * scl;
    sQ[h][r + 16][rl + 16] = s11[v] * scl;
  }
  __syncthreads();
  // softmax: lane == row
  float mx = -1e30f;
  for (int j = 0; j < 32; ++j) mx = fmaxf(mx, sQ[h][lane][j]);
  float sum = 0.0f;
  for (int j = 0; j < 32; ++j) {
    const float e = __expf(sQ[h][lane][j] - mx);
    sQ[h][lane][j] = e;
    sum += e;
  }
  const float rs = 1.0f / sum;
  for (int j = 0; j < 32; ++j) sQ[h][lane][j] *= rs;
  __syncthreads();
  v8f o00 = {}, o01 = {}, o10 = {}, o11 = {};
#pragma unroll
  for (int kk = 0; kk < 32; kk += 4) {
    const int kf = kk + kh;
    v2f a0, a1, b0, b1;
    a0.x = sQ[h][rl][kf];       a0.y = sQ[h][rl][kf + 1];
    a1.x = sQ[h][rl + 16][kf];  a1.y = sQ[h][rl + 16][kf + 1];
    b0.x = sV[h][kf][rl];       b0.y = sV[h][kf + 1][rl];
    b1.x = sV[h][kf][rl + 16];  b1.y = sV[h][kf + 1][rl + 16];
    o00 = wmma_f32(a0, b0, o00); o01 = wmma_f32(a0, b1, o01);
    o10 = wmma_f32(a1, b0, o10); o11 = wmma_f32(a1, b1, o11);
  }
  float* ob = o + (size_t)rowBase * DD + h * HD;
#pragma unroll
  for (int v = 0; v < 8; ++v) {
    const int r = rowOff + v;
    ob[(size_t)r * DD + rl]             = o00[v];
    ob[(size_t)r * DD + rl + 16]        = o01[v];
    ob[(size_t)(r + 16) * DD + rl]      = o10[v];
    ob[(size_t)(r + 16) * DD + rl + 16] = o11[v];
  }
}

// ---------------------------------------------------------------------------
// 5) Leaf Gram matrix: dense = (U U^T) * exp(log_scale) + diag; one wave/leaf
// ---------------------------------------------------------------------------
__global__ __launch_bounds__(32) void dense_kernel(
    const float* __restrict__ u, const float* __restrict__ x,
    const float* __restrict__ log_scale, float* __restrict__ out)
{
  __shared__ float sU[32][33];
  const int lane = threadIdx.x;
  const int leaf = blockIdx.x;            // b*256 + l
  const int b = leaf >> 8, l = leaf & 255;
  const float* ur = u + (size_t)leaf * LEAF * LEAF;
  for (int d = 0; d < 32; ++d) sU[lane][d] = ur[lane * 32 + d];
  __syncthreads();
  const float scale = __expf(log_scale[0]);
  const int rl = lane & 15;
  const int kh = (lane >> 4) << 1;
  v8f g00 = {}, g01 = {}, g10 = {}, g11 = {};
#pragma unroll
  for (int kk = 0; kk < 32; kk += 4) {
    const int kf = kk + kh;
    v2f a0, a1;
    a0.x = sU[rl][kf];       a0.y = sU[rl][kf + 1];
    a1.x = sU[rl + 16][kf];  a1.y = sU[rl + 16][kf + 1];
    // A fragment and B(=U^T) fragment read identical LDS locations
    g00 = wmma_f32(a0, a0, g00); g01 = wmma_f32(a0, a1, g01);
    g10 = wmma_f32(a1, a0, g10); g11 = wmma_f32(a1, a1, g11);
  }
  const int rowOff = (lane >> 4) << 3;
  float* ob = out + (size_t)b * OUT_STRIDE + (size_t)l * 1024;
#pragma unroll
  for (int v = 0; v < 8; ++v) {
    const int i = rowOff + v;
    float d00 = g00[v] * scale, d01 = g01[v] * scale;
    float d10 = g10[v] * scale, d11 = g11[v] * scale;
    if (i == rl) {  // diagonal of (i,i) and (i+16,i+16)
      const int n0 = l * 32 + i;
      const float da = x[((size_t)b * NN + n0) * 124 + 3];
      d00 += (fabsf(da) > 1e-6f) ? (1.0f / da) : 1.0f;
      const int n1 = n0 + 16;
      const float db = x[((size_t)b * NN + n1) * 124 + 3];
      d11 += (fabsf(db) > 1e-6f) ? (1.0f / db) : 1.0f;
    }
    ob[(size_t)i * 32 + rl]             = d00;
    ob[(size_t)i * 32 + rl + 16]        = d01;
    ob[(size_t)(i + 16) * 32 + rl]      = d10;
    ob[(size_t)(i + 16) * 32 + rl + 16] = d11;
  }
}

// ---------------------------------------------------------------------------
// 6) Scatter U/V (B, n_j, EXP_j) == (B, N, R_j) view into packed output
// ---------------------------------------------------------------------------
__global__ __launch_bounds__(256) void scatter_kernel(
    const float* __restrict__ src, float* __restrict__ out,
    int R, int rankOff, long sectionBase, long total)
{
  const long idx = (long)blockIdx.x * 256 + threadIdx.x;
  if (idx >= total) return;
  const int r = (int)(idx % R);
  const long nn = idx / R;
  const int n = (int)(nn % NN);
  const int b = (int)(nn / NN);
  out[(size_t)b * OUT_STRIDE + sectionBase + (size_t)n * 186 + rankOff + r] =
      src[idx];
}

// ---------------------------------------------------------------------------
extern "C" void kernel_launch(void* const* d_in, const int* in_sizes, int n_in,
                              void* d_out, int out_size, void* d_ws,
                              size_t ws_size, hipStream_t stream)
{
  (void)in_sizes; (void)n_in; (void)out_size; (void)ws_size;
  const float* x       = (const float*)d_in[0];
  const float* cen_w   = (const float*)d_in[1];
  const float* cen_b   = (const float*)d_in[2];
  const float* nbr_w   = (const float*)d_in[3];
  const float* nbr_b   = (const float*)d_in[4];
  const float* inp_w   = (const float*)d_in[5];
  const float* inp_b   = (const float*)d_in[6];
  const float* leafp_w = (const float*)d_in[7];
  const float* leafp_b = (const float*)d_in[8];
  const float* leafh_w = (const float*)d_in[9];
  const float* leafh_b = (const float*)d_in[10];
  const float* lscale  = (const float*)d_in[11];
  const float* ln1_g   = (const float*)d_in[12];
  const float* ln1_b   = (const float*)d_in[13];
  const float* qkv_w   = (const float*)d_in[14];
  const float* qkv_b   = (const float*)d_in[15];
  const float* attnp_w = (const float*)d_in[16];
  const float* attnp_b = (const float*)d_in[17];
  const float* ln2_g   = (const float*)d_in[18];
  const float* ln2_b   = (const float*)d_in[19];
  const float* mlp1_w  = (const float*)d_in[20];
  const float* mlp1_b  = (const float*)d_in[21];
  const float* mlp2_w  = (const float*)d_in[22];
  const float* mlp2_b  = (const float*)d_in[23];
  const float* ds_w    = (const float*)d_in[24];
  const float* ds_b    = (const float*)d_in[25];

  float* out = (float*)d_out;
  char* ws = (char*)d_ws;
  const size_t MB = (size_t)1 << 20;
  float* bufH   = (float*)(ws + 0 * MB);     // h current        (<=16MB)
  float* bufH2  = (float*)(ws + 16 * MB);    // h ping-pong      (<= 8MB)
  float* bufLN  = (float*)(ws + 32 * MB);    // ln out / attn o  (<=16MB)
  float* bufQKV = (float*)(ws + 48 * MB);    // qkv / h_pre      (<=48MB)
  float* bufMID = (float*)(ws + 96 * MB);    // mlp mid / hl     (<=64MB)
  float* bufU   = (float*)(ws + 160 * MB);   // u_leaf           (  4MB)
  float* bufUV  = (float*)(ws + 168 * MB);   // hu/hv tmp        (<=11MB)

  const int M0 = BB * NN;  // 32768 tokens

  // Embedding + input projection
  embed_kernel<<<M0 / 8, 256, 0, stream>>>(x, cen_w, cen_b, nbr_w, nbr_b, bufQKV);
  gemm_kernel<<<dim3(2, M0 / 64), 128, 0, stream>>>(bufQKV, inp_w, inp_b,
                                                    nullptr, bufH, M0, 128, 128, 0);
  // Leaf dense branch (uses h before the transformer stack)
  gemm_kernel<<<dim3(1, M0 / 64), 128, 0, stream>>>(bufH, leafp_w, leafp_b,
                                                    nullptr, bufMID, M0, 128, 32, 0);
  gemm_kernel<<<dim3(1, M0 / 64), 128, 0, stream>>>(bufMID, leafh_w, leafh_b,
                                                    nullptr, bufU, M0, 32, 32, 0);
  dense_kernel<<<BB * 256, 32, 0, stream>>>(bufU, x, lscale, out);

  float* h = bufH;
  float* hNext = bufH2;
  int Mj = M0;
  const int EXPJ[4] = {20, 64, 208, 656};
  const int RJ[4]   = {20, 32, 52, 82};
  const int ROFF[4] = {0, 20, 52, 104};

  for (int j = 0; j < 4; ++j) {
    // Attention block
    layernorm_kernel<<<Mj / 8, 256, 0, stream>>>(h, ln1_g + j * DD,
                                                 ln1_b + j * DD, bufLN, Mj);
    gemm_kernel<<<dim3(6, Mj / 64), 128, 0, stream>>>(
        bufLN, qkv_w + (size_t)j * DD * 384, qkv_b + j * 384, nullptr, bufQKV,
        Mj, 128, 384, 0);
    attn_kernel<<<Mj / 32, 128, 0, stream>>>(bufQKV, bufLN);
    gemm_kernel<<<dim3(2, Mj / 64), 128, 0, stream>>>(
        bufLN, attnp_w + (size_t)j * DD * DD, attnp_b + j * DD, h, h,
        Mj, 128, 128, 0);
    // MLP block (exact GELU fused in epilogue)
    layernorm_kernel<<<Mj / 8, 256, 0, stream>>>(h, ln2_g + j * DD,
                                                 ln2_b + j * DD, bufLN, Mj);
    gemm_kernel<<<dim3(8, Mj / 64), 128, 0, stream>>>(
        bufLN, mlp1_w + (size_t)j * DD * 512, mlp1_b + j * 512, nullptr, bufMID,
        Mj, 128, 512, 1);
    gemm_kernel<<<dim3(2, Mj / 64), 128, 0, stream>>>(
        bufMID, mlp2_w + (size_t)j * 512 * DD, mlp2_b + j * DD, h, h,
        Mj, 512, 128, 0);
    // U/V rank projections, scattered into the packed output
    const float* huw = (const float*)d_in[26 + 4 * j];
    const float* hub = (const float*)d_in[27 + 4 * j];
    const float* hvw = (const float*)d_in[28 + 4 * j];
    const float* hvb = (const float*)d_in[29 + 4 * j];
    const long total = (long)BB * NN * RJ[j];
    gemm_kernel<<<dim3((EXPJ[j] + 63) / 64, Mj / 64), 128, 0, stream>>>(
        h, huw, hub, nullptr, bufUV, Mj, 128, EXPJ[j], 0);
    scatter_kernel<<<(int)((total + 255) / 256), 256, 0, stream>>>(
        bufUV, out, RJ[j], ROFF[j], US_BASE, total);
    gemm_kernel<<<dim3((EXPJ[j] + 63) / 64, Mj / 64), 128, 0, stream>>>(
        h, hvw, hvb, nullptr, bufUV, Mj, 128, EXPJ[j], 0);
    scatter_kernel<<<(int)((total + 255) / 256), 256, 0, stream>>>(
        bufUV, out, RJ[j], ROFF[j], VS_BASE, total);
    // Downsample: (Mj,128) viewed as (Mj/2,256) @ ds_w -> (Mj/2,128)
    if (j < 3) {
      gemm_kernel<<<dim3(2, (Mj / 2) / 64), 128, 0, stream>>>(
          h, ds_w + (size_t)j * 256 * DD, ds_b + j * DD, nullptr, hNext,
          Mj / 2, 256, 128, 0);
      float* tmp = h; h = hNext; hNext = tmp;
      Mj /= 2;
    }
  }
}